// CAM_Module_19481971655243
// MI455X (gfx1250) — compile-verified
//
#include <hip/hip_runtime.h>
#include <hip/hip_bf16.h>
#include <stdint.h>

typedef __attribute__((ext_vector_type(2))) float v2f;
typedef __attribute__((ext_vector_type(4))) float v4f;
typedef __attribute__((ext_vector_type(8))) float v8f;

#define NB 16
#define NC 64
#define NN 65536
#define CHUNK1 1024   // N columns per workgroup in gram phase -> 64 chunks
#define KT 64         // columns per LDS stage in gram phase
#define LDS_STRIDE 68 // 64 + 4 floats pad -> conflict-free ds_load_b128
#define NT3   1024    // N columns per workgroup in output phase -> 64 blocks

__device__ __forceinline__ void atomic_add_f32(float* p, float v) {
    __hip_atomic_fetch_add(p, v, __ATOMIC_RELAXED, __HIP_MEMORY_SCOPE_AGENT);
}

__device__ __forceinline__ v8f wmma_f32(v2f a, v2f b, v8f c) {
    return __builtin_amdgcn_wmma_f32_16x16x4_f32(
        /*neg_a=*/false, a, /*neg_b=*/false, b,
        /*c_mod=*/(short)0, c, /*reuse_a=*/false, /*reuse_b=*/false);
}

__device__ __forceinline__ void async_load_lds_b128(uint32_t lds_byte_off,
                                                    const float* gaddr) {
    asm volatile("global_load_async_to_lds_b128 %0, %1, off"
                 :: "v"(lds_byte_off), "v"((uint64_t)(uintptr_t)gaddr)
                 : "memory");
}

__device__ __forceinline__ void wait_async0() {
    asm volatile("s_wait_asynccnt 0" ::: "memory");
}

// ---------------------------------------------------------------------------
// Kernel 0: zero the energy accumulator (16*64*64 floats in d_ws)
// ---------------------------------------------------------------------------
__global__ __launch_bounds__(1024) void zero_kernel(float* __restrict__ p, int n) {
    int i = blockIdx.x * blockDim.x + threadIdx.x;
    if (i < n) p[i] = 0.0f;
}

// ---------------------------------------------------------------------------
// Kernel 1: energy[b] += X_b (64 x CHUNK1 slice) * X_b^T via fp32 WMMA.
// grid = (N/CHUNK1, B), block = 128 (4 waves). Wave w owns row tile i0=16w.
// Global->LDS staging uses coalesced async copies (each row = 2 cachelines),
// double-buffered under ASYNCcnt; WMMA fragments come from LDS (bank-conflict
// free via 68-float row stride). Per 8 columns: one ds_load_b128 per fragment
// feeds two K=4 WMMAs (column->K-slot permutation identical for A and B).
// ---------------------------------------------------------------------------
__global__ __launch_bounds__(128) void gram_kernel(const float* __restrict__ x,
                                                   float* __restrict__ energy) {
    extern __shared__ float smem[];   // [2][64][LDS_STRIDE]
    const int tid   = threadIdx.x;
    const int batch = blockIdx.y;
    const int wave  = tid >> 5;
    const int lane  = tid & 31;
    const int h     = lane >> 4;      // lane half: K-slot group
    const int l     = lane & 15;      // row within tile
    const int i0    = wave * 16;

    const float* __restrict__ xb = x + (size_t)batch * NC * NN;
    const int nbase = blockIdx.x * CHUNK1;

    // LDS byte offset of smem: low 32 bits of the generic address (flat rule).
    const uint32_t smem_base = (uint32_t)(uintptr_t)smem;
    const uint32_t buf_bytes = (uint32_t)(NC * LDS_STRIDE * 4);

    // Copy mapping: 128 threads * 16B = 2KB = 8 rows (KT*4 = 256B/row) per issue.
    const int crow = tid >> 4;          // 0..7
    const int ccol = (tid & 15) * 4;    // float offset within row: 0..60
    const uint32_t lds_wr = smem_base + (uint32_t)((crow * LDS_STRIDE + ccol) * 4);
    const float* gsrc = xb + (size_t)crow * NN + ccol;

    auto stage_copy = [&](int n0, int buf) {
        const uint32_t lbase = lds_wr + (uint32_t)buf * buf_bytes;
        const float* g = gsrc + n0;
#pragma unroll
        for (int i = 0; i < 8; ++i) {
            async_load_lds_b128(lbase + (uint32_t)(i * 8 * LDS_STRIDE * 4),
                                g + (size_t)(i * 8) * NN);
        }
    };

    v8f c0 = {0,0,0,0,0,0,0,0};
    v8f c1 = {0,0,0,0,0,0,0,0};
    v8f c2 = {0,0,0,0,0,0,0,0};
    v8f c3 = {0,0,0,0,0,0,0,0};

    stage_copy(nbase, 0);
    wait_async0();
    __syncthreads();

    const int S = CHUNK1 / KT;   // stages
    for (int k = 0; k < S; ++k) {
        if (k + 1 < S) stage_copy(nbase + (k + 1) * KT, (k + 1) & 1);

        const float* base = smem + (size_t)(k & 1) * NC * LDS_STRIDE;
        const float* ra  = base + (i0 + l) * LDS_STRIDE + 4 * h;
        const float* rb0 = base + ( 0 + l) * LDS_STRIDE + 4 * h;
        const float* rb1 = base + (16 + l) * LDS_STRIDE + 4 * h;
        const float* rb2 = base + (32 + l) * LDS_STRIDE + 4 * h;
        const float* rb3 = base + (48 + l) * LDS_STRIDE + 4 * h;
#pragma unroll
        for (int it = 0; it < KT / 8; ++it) {
            const int cc = it * 8;
            v4f av = *(const v4f*)(ra + cc);
            v2f a0; a0.x = av.x; a0.y = av.y;
            v2f a1; a1.x = av.z; a1.y = av.w;

            v4f bv = *(const v4f*)(rb0 + cc);
            v2f b0; b0.x = bv.x; b0.y = bv.y;
            v2f b1; b1.x = bv.z; b1.y = bv.w;
            c0 = wmma_f32(a0, b0, c0);
            c0 = wmma_f32(a1, b1, c0);

            bv = *(const v4f*)(rb1 + cc);
            b0.x = bv.x; b0.y = bv.y; b1.x = bv.z; b1.y = bv.w;
            c1 = wmma_f32(a0, b0, c1);
            c1 = wmma_f32(a1, b1, c1);

            bv = *(const v4f*)(rb2 + cc);
            b0.x = bv.x; b0.y = bv.y; b1.x = bv.z; b1.y = bv.w;
            c2 = wmma_f32(a0, b0, c2);
            c2 = wmma_f32(a1, b1, c2);

            bv = *(const v4f*)(rb3 + cc);
            b0.x = bv.x; b0.y = bv.y; b1.x = bv.z; b1.y = bv.w;
            c3 = wmma_f32(a0, b0, c3);
            c3 = wmma_f32(a1, b1, c3);
        }

        wait_async0();        // next stage's copies have landed
        __syncthreads();      // and everyone is done reading current buffer
    }

    // C/D layout: VGPR v holds element (M = v + 8h, N = l) of the 16x16 tile.
    float* eb = energy + (size_t)batch * NC * NC;
#pragma unroll
    for (int v = 0; v < 8; ++v) {
        const int row = i0 + v + 8 * h;
        atomic_add_f32(&eb[row * NC +  0 + l], c0[v]);
        atomic_add_f32(&eb[row * NC + 16 + l], c1[v]);
        atomic_add_f32(&eb[row * NC + 32 + l], c2[v]);
        atomic_add_f32(&eb[row * NC + 48 + l], c3[v]);
    }
}

// ---------------------------------------------------------------------------
// Kernel 2: attention = softmax(rowmax - energy) = exp(emin - e) / sum
// grid = (B), block = 64: one thread per (b, row).
// ---------------------------------------------------------------------------
__global__ __launch_bounds__(64) void softmax_kernel(const float* __restrict__ energy,
                                                     float* __restrict__ att) {
    const int b = blockIdx.x;
    const int row = threadIdx.x;
    const float* e = energy + ((size_t)b * NC + row) * NC;
    float* a = att + ((size_t)b * NC + row) * NC;

    float emin = 3.402823466e+38f;
#pragma unroll 8
    for (int j = 0; j < NC; ++j) emin = fminf(emin, e[j]);

    float s = 0.0f;
#pragma unroll 8
    for (int j = 0; j < NC; ++j) {
        float p = __expf(emin - e[j]);
        a[j] = p;
        s += p;
    }
    const float inv = 1.0f / s;
#pragma unroll 8
    for (int j = 0; j < NC; ++j) a[j] *= inv;
}

// ---------------------------------------------------------------------------
// Kernel 3: out = gamma * (att @ X) + X, fused epilogue.
// grid = (N/NT3, B), block = 128. Wave w owns row tile i0 = 16w; preloads its
// 16x64 attention strip as 16 WMMA A-fragments, then marches 16-column tiles.
// B loads are coalesced (two full 64B segments per b32 load).
// ---------------------------------------------------------------------------
__global__ __launch_bounds__(128) void attn_out_kernel(const float* __restrict__ x,
                                                       const float* __restrict__ att,
                                                       const float* __restrict__ gamma,
                                                       float* __restrict__ out) {
    const int batch = blockIdx.y;
    const int wave  = threadIdx.x >> 5;
    const int lane  = threadIdx.x & 31;
    const int h     = lane >> 4;
    const int l     = lane & 15;
    const int i0    = wave * 16;

    const float* __restrict__ xb = x   + (size_t)batch * NC * NN;
    float* __restrict__       ob = out + (size_t)batch * NC * NN;
    const float* __restrict__ ab = att + (size_t)batch * NC * NC;
    const float g = gamma[0];

    // A-fragments: att[i0+l, 4c + 2h .. +1]  (16 chunks cover K = 0..63)
    v2f aF[16];
#pragma unroll
    for (int c = 0; c < 16; ++c)
        aF[c] = *(const v2f*)(ab + (i0 + l) * NC + 4 * c + 2 * h);

    const int nbase = blockIdx.x * NT3;
    for (int t = 0; t < NT3 / 16; ++t) {
        const int col = nbase + t * 16 + l;
        v8f acc = {0,0,0,0,0,0,0,0};
#pragma unroll
        for (int c = 0; c < 16; ++c) {
            const int j0 = 4 * c + 2 * h;
            v2f b;
            b.x = xb[(size_t)(j0 + 0) * NN + col];
            b.y = xb[(size_t)(j0 + 1) * NN + col];
            acc = wmma_f32(aF[c], b, acc);
        }
#pragma unroll
        for (int v = 0; v < 8; ++v) {
            const size_t idx = (size_t)(i0 + v + 8 * h) * NN + col;
            ob[idx] = g * acc[v] + xb[idx];
        }
    }
}

// ---------------------------------------------------------------------------
extern "C" void kernel_launch(void* const* d_in, const int* in_sizes, int n_in,
                              void* d_out, int out_size, void* d_ws, size_t ws_size,
                              hipStream_t stream) {
    const float* x     = (const float*)d_in[0];
    const float* gamma = (const float*)d_in[1];
    float* out = (float*)d_out;

    float* energy = (float*)d_ws;                  // 16*64*64 floats (256 KB)
    float* att    = energy + (size_t)NB * NC * NC; // 16*64*64 floats (256 KB)

    const int n_energy = NB * NC * NC;
    zero_kernel<<<(n_energy + 1023) / 1024, 1024, 0, stream>>>(energy, n_energy);

    const size_t lds_bytes = 2u * NC * LDS_STRIDE * sizeof(float);
    gram_kernel<<<dim3(NN / CHUNK1, NB), 128, lds_bytes, stream>>>(x, energy);

    softmax_kernel<<<NB, NC, 0, stream>>>(energy, att);

    attn_out_kernel<<<dim3(NN / NT3, NB), 128, 0, stream>>>(x, att, gamma, out);
}